// VONet_48180943127023
// MI455X (gfx1250) — compile-verified
//
#include <hip/hip_runtime.h>
#include <hip/hip_bf16.h>

typedef __attribute__((ext_vector_type(2))) float v2f;
typedef __attribute__((ext_vector_type(8))) float v8f;

#define C_CH 128
#define H0 120
#define W0 160
#define H1 30
#define W1 40
#define PP 9          // 3x3 patch positions

// ---------------------------------------------------------------------------
// Kernel 1: 4x4 average pool of fmap -> pooled pyramid level in d_ws
// fmap: [16][128][120][160]  ->  pooled: [16][128][30][40]
// ---------------------------------------------------------------------------
__global__ __launch_bounds__(256) void pool4_kernel(const float* __restrict__ fmap,
                                                    float* __restrict__ pooled,
                                                    int total)
{
    int idx = blockIdx.x * blockDim.x + threadIdx.x;
    if (idx >= total) return;
    int wo = idx % W1;
    int t  = idx / W1;
    int ho = t % H1;
    int nc = t / H1;
    const float* src = fmap + (size_t)nc * (H0 * W0) + (4 * ho) * W0 + 4 * wo;
    float s = 0.f;
#pragma unroll
    for (int r = 0; r < 4; ++r) {
        float4 v = *(const float4*)(src + r * W0);
        s += v.x + v.y + v.z + v.w;
    }
    pooled[idx] = s * (1.0f / 16.0f);
}

// ---------------------------------------------------------------------------
// Kernel 2: correlation lookup. One workgroup per edge e.
// 288 threads = 9 waves; wave w handles patch position p = w for both levels.
// Per (e,p,level): D[8][8] = window(64 pos) x g(128) via fp32 WMMA 16x16x4,
// then 7x7 bilinear blend with validity masks.
// ---------------------------------------------------------------------------
__global__ __launch_bounds__(288) void corr_kernel(
    const float* __restrict__ fmap,    // [16][128][120][160]
    const float* __restrict__ pooled,  // [16][128][30][40]
    const float* __restrict__ gmap,    // [M][128][3][3]
    const float* __restrict__ coords,  // [E][2][3][3]
    const int*   __restrict__ ii,
    const int*   __restrict__ jj,
    float* __restrict__ out)           // [E][3][3][7][7][2]
{
    __shared__ float g_sh[PP * C_CH];  // g_sh[p*128 + c]
    __shared__ float D_sh[PP * 64];    // per-wave 8x8 dot grid

    const int e    = blockIdx.x;
    const int tid  = threadIdx.x;
    const int wave = tid >> 5;         // == p (9 waves)
    const int lane = tid & 31;
    const int p    = wave;

    const int ii_e = ii[e];
    const int jj_e = jj[e];

    // Stage g: gmap[ii_e][c][p] -> g_sh[p*128 + c]  (transpose for b64 reads)
    const float* gsrc = gmap + (size_t)ii_e * (C_CH * PP);
    for (int q = tid; q < PP * C_CH; q += 288) {
        int pq = q >> 7;           // p
        int cq = q & 127;          // c
        g_sh[q] = gsrc[cq * PP + pq];
    }
    __syncthreads();

    const float xf = coords[(size_t)e * 18 + p];       // x at level 0
    const float yf = coords[(size_t)e * 18 + 9 + p];   // y at level 0

    const int row  = lane & 15;   // A-matrix M row for this lane
    const int half = lane >> 4;   // K pair selector (0 -> K0,K1 ; 1 -> K2,K3)
    const float* gp = &g_sh[p * C_CH];
    float* D = &D_sh[wave * 64];

#pragma unroll
    for (int l = 0; l < 2; ++l) {
        // compile-time level constants after unroll
        const int Hl = l ? H1 : H0;
        const int Wl = l ? W1 : W0;
        const int HW = Hl * Wl;
        const float x = l ? (xf * 0.25f) : xf;
        const float y = l ? (yf * 0.25f) : yf;
        const float fx = floorf(x), fy = floorf(y);
        const float wx = x - fx,   wy = y - fy;        // constant over window
        const int xb = (int)fx - 3;                    // window base (d=-3 corner)
        const int yb = (int)fy - 3;
        // uniform (scalar) frame base; lane-varying part kept as 32-bit offset
        const float* fframe = (l ? pooled : fmap) + (size_t)jj_e * C_CH * HW;

        // ---- D[64] = F_window[64 x 128] * g[128]  via V_WMMA_F32_16X16X4_F32
        // B is broadcast across all 16 N columns -> every lane holds D values.
        for (int m = 0; m < 4; ++m) {
            int pos = 16 * m + row;          // position index 0..63
            int i = pos >> 3, j = pos & 7;   // window row / col
            int yc = min(max(yb + i, 0), Hl - 1);
            int xc = min(max(xb + j, 0), Wl - 1);
            // lane-varying 32-bit element offset (channel term is constant)
            const int base = yc * Wl + xc + 2 * half * HW;
            v8f acc = {};
#pragma unroll
            for (int t = 0; t < 32; ++t) {
                v2f a, b;
                a.x = fframe[base + t * 4 * HW];       // F[c = 4t + 2*half    , pos]
                a.y = fframe[base + t * 4 * HW + HW];  // F[c = 4t + 2*half + 1, pos]
                int gc = 4 * t + 2 * half;
                b.x = gp[gc];                // broadcast B: g[c, p] for all N
                b.y = gp[gc + 1];
                acc = __builtin_amdgcn_wmma_f32_16x16x4_f32(
                    false, a, false, b, (short)0, acc, false, false);
            }
            // C layout: VGPR r, lanes 0-15 -> M=r, lanes 16-31 -> M=8+r.
            // All N columns identical; lanes 0 and 16 publish D.
            if (row == 0) {
#pragma unroll
                for (int r = 0; r < 8; ++r)
                    D[16 * m + 8 * half + r] = acc[r];
            }
        }
        __syncthreads();

        // ---- bilinear blend with reference-exact validity masking
        float* op = out + ((size_t)e * PP + p) * 49 * 2 + l;
        for (int k = lane; k < 49; k += 32) {
            int ky = k / 7, kx = k % 7;
            int xl = xb + kx, xr = xl + 1;
            int yt = yb + ky, yd = yt + 1;
            float vx0 = (xl >= 0 && xl < Wl) ? 1.f : 0.f;
            float vx1 = (xr >= 0 && xr < Wl) ? 1.f : 0.f;
            float vy0 = (yt >= 0 && yt < Hl) ? 1.f : 0.f;
            float vy1 = (yd >= 0 && yd < Hl) ? 1.f : 0.f;
            float d00 = D[ky * 8 + kx];
            float d01 = D[ky * 8 + kx + 1];
            float d10 = D[(ky + 1) * 8 + kx];
            float d11 = D[(ky + 1) * 8 + kx + 1];
            float r = (1.f - wy) * (1.f - wx) * vy0 * vx0 * d00
                    + (1.f - wy) * wx         * vy0 * vx1 * d01
                    + wy         * (1.f - wx) * vy1 * vx0 * d10
                    + wy         * wx         * vy1 * vx1 * d11;
            op[2 * k] = r;
        }
        __syncthreads();
    }
}

// ---------------------------------------------------------------------------
extern "C" void kernel_launch(void* const* d_in, const int* in_sizes, int n_in,
                              void* d_out, int out_size, void* d_ws, size_t ws_size,
                              hipStream_t stream)
{
    const float* fmap   = (const float*)d_in[0];
    const float* gmap   = (const float*)d_in[1];
    const float* coords = (const float*)d_in[2];
    const int*   ii     = (const int*)d_in[3];
    const int*   jj     = (const int*)d_in[4];
    float* out = (float*)d_out;
    float* pooled = (float*)d_ws;   // 16*128*30*40 floats = 9.83 MB

    const int E = in_sizes[3];      // 2048

    int pool_total = 16 * C_CH * H1 * W1;
    pool4_kernel<<<(pool_total + 255) / 256, 256, 0, stream>>>(fmap, pooled, pool_total);
    corr_kernel<<<E, 288, 0, stream>>>(fmap, pooled, gmap, coords, ii, jj, out);
}